// Rank_IGR_Loss_FAST_13967233646996
// MI455X (gfx1250) — compile-verified
//
#include <hip/hip_runtime.h>
#include <hip/hip_bf16.h>
#include <math.h>

#define KPOS 16
#define GAMMA 3.0f
#define NPAIR 120.0f

typedef __attribute__((ext_vector_type(2))) float v2f;
typedef __attribute__((ext_vector_type(8))) float v8f;

// One workgroup per sample. 256 threads: vectorized coalesced label scan;
// lanes 0-15 do branch-free rank sorts; wave 0 does two rank-1 WMMA outer
// products (V_WMMA_F32_16X16X4_F32) for the pairwise ranking losses.
__global__ __launch_bounds__(256) void rank_igr_sample_kernel(
    const float* __restrict__ cls,     // [B,N,2]
    const int*   __restrict__ label,   // [B,N]
    const float* __restrict__ boxes,   // [B,4,N]
    const float* __restrict__ tgt,     // [B,4]
    float* __restrict__ ws,            // [2*B]  ws[b]=l1_b, ws[B+b]=l2_b
    int B, int N)
{
    const int b   = blockIdx.x;
    const int tid = threadIdx.x;

    __shared__ int   s_count;
    __shared__ int   s_raw[KPOS];            // positives, arbitrary order
    __shared__ int   s_idx[KPOS];            // positives, ascending
    __shared__ float s_prob[KPOS], s_iou[KPOS];
    __shared__ float s_p[KPOS], s_q[KPOS];   // rank-sorted value arrays
    __shared__ float s_cp, s_cq;             // centers for overflow-safe exp split
    __shared__ float s_part1[32], s_part2[32];

    if (tid == 0) s_count = 0;
    __syncthreads();

    // ---- Phase A: coalesced 128-bit label scan ----
    const int* lab = label + (size_t)b * N;
    const int  N4  = N >> 2;
    const int4* lab4 = (const int4*)lab;
    for (int i = tid; i < N4; i += blockDim.x) {
        const int4 v = lab4[i];
        const int base = i << 2;
        if (v.x == 1) { int s = atomicAdd(&s_count, 1); if (s < KPOS) s_raw[s] = base;     }
        if (v.y == 1) { int s = atomicAdd(&s_count, 1); if (s < KPOS) s_raw[s] = base + 1; }
        if (v.z == 1) { int s = atomicAdd(&s_count, 1); if (s < KPOS) s_raw[s] = base + 2; }
        if (v.w == 1) { int s = atomicAdd(&s_count, 1); if (s < KPOS) s_raw[s] = base + 3; }
    }
    for (int n = (N4 << 2) + tid; n < N; n += blockDim.x) {   // tail (N not /4)
        if (lab[n] == 1) { int s = atomicAdd(&s_count, 1); if (s < KPOS) s_raw[s] = n; }
    }
    __syncthreads();

    // ---- branch-free rank sort of indices (ascending, all distinct) ----
    if (tid < KPOS) {
        const int v = s_raw[tid];
        int r = 0;
        #pragma unroll
        for (int j = 0; j < KPOS; ++j) r += (s_raw[j] < v) ? 1 : 0;
        s_idx[r] = v;
    }
    __syncthreads();

    // ---- Phase B: gather prob & IoU at the 16 positives ----
    if (tid < KPOS) {
        const int n = s_idx[tid];
        const float prob = expf(cls[((size_t)b * N + n) * 2 + 1]);
        const float* bb = boxes + (size_t)b * 4 * N;
        const float x1 = bb[n],               y1 = bb[(size_t)N + n];
        const float x2 = bb[2*(size_t)N + n], y2 = bb[3*(size_t)N + n];
        const float tx1 = tgt[b*4+0], ty1 = tgt[b*4+1];
        const float tx2 = tgt[b*4+2], ty2 = tgt[b*4+3];
        float ww = fmaxf(fminf(tx2, x2) - fmaxf(tx1, x1), 0.0f);
        float hh = fmaxf(fminf(ty2, y2) - fmaxf(ty1, y1), 0.0f);
        float inter = ww * hh;
        float uni = (x2 - x1) * (y2 - y1) + (tx2 - tx1) * (ty2 - ty1) - inter;
        s_prob[tid] = prob;
        s_iou[tid]  = inter / uni;
    }
    __syncthreads();

    // ---- branch-free stable descending rank sorts (matches jnp.argsort(-x)) ----
    // position of element k = #{j: key_j > key_k} + #{j<k: key_j == key_k}
    if (tid < KPOS) {
        const float kI = s_iou[tid], kP = s_prob[tid];
        int r1 = 0, r2 = 0;
        #pragma unroll
        for (int j = 0; j < KPOS; ++j) {
            const float ij = s_iou[j], pj = s_prob[j];
            r1 += ((ij > kI) || (ij == kI && j < tid)) ? 1 : 0;
            r2 += ((pj > kP) || (pj == kP && j < tid)) ? 1 : 0;
        }
        s_p[r1] = s_prob[tid];   // p: prob ordered by descending IoU
        s_q[r2] = s_iou[tid];    // q: IoU ordered by descending prob
    }
    __syncthreads();

    if (tid == 0) {   // centers (order-independent)
        float mnp = s_prob[0], mxp = s_prob[0], mnq = s_iou[0], mxq = s_iou[0];
        for (int i = 1; i < KPOS; ++i) {
            mnp = fminf(mnp, s_prob[i]); mxp = fmaxf(mxp, s_prob[i]);
            mnq = fminf(mnq, s_iou[i]);  mxq = fmaxf(mxq, s_iou[i]);
        }
        s_cp = 0.5f * (mnp + mxp);
        s_cq = 0.5f * (mnq + mxq);
    }
    __syncthreads();

    // ---- Phase C: wave 0 only — two rank-1 outer products via WMMA f32 16x16x4
    // exp(-g*(v_i - v_j)) = exp(-g*(v_i-c)) * exp(+g*(v_j-c)); D[m][n] = a_m * b_n.
    // Only the K=0 slice of A (column, lanes 0-15 of v0) and B (row, lanes 0-15
    // of v0) is populated; all other A/B registers are zero.
    if (tid < 32) {
        const int  lane = tid;
        const int  nn   = lane & 15;
        const bool lo   = lane < 16;

        const float cp = s_cp;
        v2f Av = {0.0f, 0.0f}, Bv = {0.0f, 0.0f};
        if (lo) {
            Av.x = expf(-GAMMA * (s_p[nn] - cp));   // A[M=nn][K=0]
            Bv.x = expf( GAMMA * (s_p[nn] - cp));   // B[K=0][N=nn]
        }
        v8f C1 = {};
        C1 = __builtin_amdgcn_wmma_f32_16x16x4_f32(
                 false, Av, false, Bv, (short)0, C1, false, false);
        float sum1 = 0.0f;
        #pragma unroll
        for (int r = 0; r < 8; ++r) {               // D: VGPR r -> M=r (lanes>=16: M=r+8)
            const int m = r + (lo ? 0 : 8);
            if (m < nn) sum1 += C1[r];              // strict upper triangle i<j
        }

        const float cq = s_cq;
        v2f A2 = {0.0f, 0.0f}, B2 = {0.0f, 0.0f};
        if (lo) {
            A2.x = expf(-GAMMA * (s_q[nn] - cq));
            B2.x = expf( GAMMA * (s_q[nn] - cq));
        }
        v8f C2 = {};
        C2 = __builtin_amdgcn_wmma_f32_16x16x4_f32(
                 false, A2, false, B2, (short)0, C2, false, false);
        float sum2 = 0.0f;
        #pragma unroll
        for (int r = 0; r < 8; ++r) {
            const int m = r + (lo ? 0 : 8);
            if (m < nn) sum2 += C2[r];
        }

        s_part1[lane] = sum1;
        s_part2[lane] = sum2;
    }
    __syncthreads();

    if (tid == 0) {   // deterministic fixed-order reduction across 32 lanes
        float l1 = 0.0f, l2 = 0.0f;
        for (int i = 0; i < 32; ++i) { l1 += s_part1[i]; l2 += s_part2[i]; }
        ws[b]     = l1 / NPAIR;
        ws[B + b] = l2 / NPAIR;
    }
}

// Deterministic final mean over B samples -> d_out[0]=l1.mean, d_out[1]=l2.mean
__global__ void rank_igr_reduce_kernel(const float* __restrict__ ws,
                                       float* __restrict__ out, int B)
{
    if (blockIdx.x == 0 && threadIdx.x == 0) {
        float s1 = 0.0f, s2 = 0.0f;
        for (int i = 0; i < B; ++i) { s1 += ws[i]; s2 += ws[B + i]; }
        out[0] = s1 / (float)B;
        out[1] = s2 / (float)B;
    }
}

extern "C" void kernel_launch(void* const* d_in, const int* in_sizes, int n_in,
                              void* d_out, int out_size, void* d_ws, size_t ws_size,
                              hipStream_t stream) {
    const float* cls   = (const float*)d_in[0];  // [B,N,2] f32
    const int*   label = (const int*)d_in[1];    // [B,N] i32
    const float* boxes = (const float*)d_in[2];  // [B,4,N] f32
    const float* tgt   = (const float*)d_in[3];  // [B,4] f32

    const int B = in_sizes[3] / 4;
    const int N = in_sizes[1] / B;

    float* ws = (float*)d_ws;   // needs 2*B floats (2 KB) of scratch

    rank_igr_sample_kernel<<<B, 256, 0, stream>>>(cls, label, boxes, tgt, ws, B, N);
    rank_igr_reduce_kernel<<<1, 32, 0, stream>>>(ws, (float*)d_out, B);
}